// Net_88871463289070
// MI455X (gfx1250) — compile-verified
//
#include <hip/hip_runtime.h>
#include <hip/hip_bf16.h>

typedef float v2f __attribute__((ext_vector_type(2)));
typedef float v8f __attribute__((ext_vector_type(8)));

#define KEY_MIN ((int)0x80000000)   // < key(-inf); decodes to NaN (unreachable)
#define KEY_MAX ((int)0x7FFFFFFF)   // > key(+inf); decodes to NaN (unreachable)

// Total-order key: signed-int compare on keys == float compare (no NaN inputs).
__device__ __forceinline__ int f2key(float v) {
    int b = __float_as_int(v);
    return b ^ ((b >> 31) & 0x7FFFFFFF);
}
__device__ __forceinline__ float key2f(int k) {
    return __int_as_float(k ^ ((k >> 31) & 0x7FFFFFFF));
}

// Per-node packed stats record: {sum.f32, cnt.u32, maxkey.i32, minkey.i32}
// One 16B record per node -> single b128 gather, atomics hit one L2 sector.

// ---------------------------------------------------------------------------
// Kernel A: init per-node stats
// ---------------------------------------------------------------------------
__global__ void init_nodes(int* __restrict__ stats, int n) {
    int i = blockIdx.x * blockDim.x + threadIdx.x;
    if (i < n) {
        int4 z;
        z.x = 0;            // sum = 0.0f
        z.y = 0;            // cnt = 0
        z.z = KEY_MIN;      // max key
        z.w = KEY_MAX;      // min key
        ((int4*)stats)[i] = z;   // global_store_b128
    }
}

__device__ __forceinline__ void scatter_atomics(int* __restrict__ stats,
                                                int node, float v) {
    int* s = stats + (size_t)node * 4;
    atomicAdd((float*)s, v);                  // global_atomic_add_f32
    atomicAdd((unsigned*)(s + 1), 1u);        // global_atomic_add_u32
    int k = f2key(v);
    atomicMax(s + 2, k);                      // global_atomic_max_i32 (branchless)
    atomicMin(s + 3, k);                      // global_atomic_min_i32 (branchless)
}

// ---------------------------------------------------------------------------
// Kernel B: ev = edge_attr @ w + b via v_wmma_f32_16x16x4_f32 (16 edges/tile,
// K=16 as 4 chained WMMAs; B = w replicated across all 16 N columns so every
// D column equals ev), then undirected scatter with native atomics.
// Lanes 0-15 handle dst endpoints, lanes 16-31 handle src endpoints.
// ---------------------------------------------------------------------------
__global__ void edge_mlp_scatter(const float* __restrict__ edge_attr,
                                 const int* __restrict__ adjs,
                                 const float* __restrict__ lin_e_w,
                                 const float* __restrict__ lin_e_b,
                                 float* __restrict__ ev,
                                 int* __restrict__ stats,
                                 int E, int ntiles) {
    const int lane   = threadIdx.x & 31;
    const int wib    = threadIdx.x >> 5;
    const int wpb    = blockDim.x >> 5;
    const int gwave  = blockIdx.x * wpb + wib;
    const int nwaves = gridDim.x * wpb;
    const bool lo    = lane < 16;

    // B operand: 4x16 f32, all columns = w[4i..4i+3] for chain step i.
    // Layout (K-half split across lane halves, consecutive K across VGPRs):
    //   VGPR0: lanes0-15 -> K=4i+0, lanes16-31 -> K=4i+2 ; VGPR1: +1 / +3.
    float bx[4], by[4];
#pragma unroll
    for (int i = 0; i < 4; ++i) {
        float w0 = lin_e_w[4 * i + 0], w1 = lin_e_w[4 * i + 1];
        float w2 = lin_e_w[4 * i + 2], w3 = lin_e_w[4 * i + 3];
        bx[i] = lo ? w0 : w2;
        by[i] = lo ? w1 : w3;
    }
    const float bias    = lin_e_b[0];
    const int   m       = lane & 15;       // A-matrix row owned by this lane
    const int   colbase = lo ? 0 : 2;      // A layout: lane halves split K pairs
    const int   j       = lane & 7;
    const int   srcLane = (lane < 8) ? lane
                        : (lane < 16) ? (lane + 16)
                        : (lane < 24) ? (lane - 16) : lane;
    const int* __restrict__ dstIdx = adjs + E;

    for (int t = gwave; t < ntiles; t += nwaves) {
        const int e0 = t << 4;
        const float* rowp = edge_attr + (size_t)(e0 + m) * 16 + colbase;

        // prefetch this wave's next tile (read-once stream)
        __builtin_prefetch(rowp + (size_t)nwaves * 16 * 16, 0, 0);

        v8f acc = {};
#pragma unroll
        for (int i = 0; i < 4; ++i) {
            v2f a, b;
            a.x = __builtin_nontemporal_load(rowp + 4 * i + 0);  // A[m, K]
            a.y = __builtin_nontemporal_load(rowp + 4 * i + 1);  // A[m, K+1]
            b.x = bx[i];
            b.y = by[i];
            acc = __builtin_amdgcn_wmma_f32_16x16x4_f32(
                false, a, false, b, (short)0, acc, false, false);
        }
        // Lane L<16 holds acc[k] = ev[e0+k]; lane L>=16 holds ev[e0+8+k].
        float a0 = (j & 1) ? acc[1] : acc[0];
        float a1 = (j & 1) ? acc[3] : acc[2];
        float a2 = (j & 1) ? acc[5] : acc[4];
        float a3 = (j & 1) ? acc[7] : acc[6];
        float b0 = (j & 2) ? a1 : a0;
        float b1 = (j & 2) ? a3 : a2;
        float t8 = (j & 4) ? b1 : b0;
        // Redistribute so lane L holds ev[e0 + (L % 16)].
        float evv = __shfl(t8, srcLane, 32) + bias;

        if (lo) ev[e0 + lane] = evv;

        int node = lo ? dstIdx[e0 + lane] : adjs[e0 + lane - 16];
        scatter_atomics(stats, node, evv);
    }
}

// ---------------------------------------------------------------------------
// Scalar tail for E not a multiple of 16 (E=3.2M is, but stay robust).
// ---------------------------------------------------------------------------
__global__ void edge_tail(const float* __restrict__ edge_attr,
                          const int* __restrict__ adjs,
                          const float* __restrict__ lin_e_w,
                          const float* __restrict__ lin_e_b,
                          float* __restrict__ ev,
                          int* __restrict__ stats,
                          int start, int E) {
    int e = start + blockIdx.x * blockDim.x + threadIdx.x;
    if (e >= E) return;
    const float* row = edge_attr + (size_t)e * 16;
    float s = lin_e_b[0];
#pragma unroll
    for (int k = 0; k < 16; ++k) s += row[k] * lin_e_w[k];
    ev[e] = s;
    scatter_atomics(stats, adjs[E + e], s);  // dst
    scatter_atomics(stats, adjs[e], s);      // src
}

// ---------------------------------------------------------------------------
// Kernel C: gather packed node stats at dst[e] (single b128), inverse-clean,
// affine combine, NT store. cnt[dst[e]] >= 1 always (the edge itself
// contributes to its dst segment), so no empty-segment masking is needed.
// ---------------------------------------------------------------------------
__global__ void finalize(const float* __restrict__ ev,
                         const int* __restrict__ adjs,
                         const int* __restrict__ stats,
                         const float* __restrict__ aggr2_w,
                         const float* __restrict__ aggr2_b,
                         const float* __restrict__ lin_l_w,
                         const float* __restrict__ lin_l_b,
                         float* __restrict__ out, int E) {
    int e = blockIdx.x * blockDim.x + threadIdx.x;
    if (e >= E) return;
    const float INF = __builtin_inff();

    int d = adjs[E + e];                      // gather at dst
    int4 st = ((const int4*)stats)[d];        // one global_load_b128
    float fsum  = __int_as_float(st.x);
    float fcnt  = (float)(unsigned)st.y;
    float fmx   = key2f(st.z);
    float fmn   = key2f(st.w);
    float fmean = fsum / fmaxf(fcnt, 1.0f);

    auto ic = [INF](float v) {
        float r = 1.0f / v;
        return (r != r || r == INF) ? 1.0f : r;   // NaN or +inf -> 1.0
    };
    float nc2 = aggr2_w[0] * ic(fmx) + aggr2_w[1] * ic(fmean) +
                aggr2_w[2] * ic(fmn) + aggr2_w[3] * ic(fsum) + aggr2_b[0];

    float evv = ev[e];
    float ec  = (evv != evv || evv == INF) ? 1.0f : evv;
    float o   = ec * lin_l_w[0] + lin_l_b[0] + nc2 * ec;
    if (o == INF) o = 1.0f;
    __builtin_nontemporal_store(o, out + e);
}

// ---------------------------------------------------------------------------
extern "C" void kernel_launch(void* const* d_in, const int* in_sizes, int n_in,
                              void* d_out, int out_size, void* d_ws, size_t ws_size,
                              hipStream_t stream) {
    (void)n_in; (void)out_size; (void)ws_size;
    // setup_inputs order: x, adjs, edge_attr, lin_e_w, lin_e_b, aggr2_w,
    //                     aggr2_b, lin_l_w, lin_l_b
    const int*   adjs      = (const int*)d_in[1];
    const float* edge_attr = (const float*)d_in[2];
    const float* lin_e_w   = (const float*)d_in[3];
    const float* lin_e_b   = (const float*)d_in[4];
    const float* aggr2_w   = (const float*)d_in[5];
    const float* aggr2_b   = (const float*)d_in[6];
    const float* lin_l_w   = (const float*)d_in[7];
    const float* lin_l_b   = (const float*)d_in[8];
    float* out = (float*)d_out;

    const int E = in_sizes[1] / 2;  // adjs is [2, E]
    const int N = in_sizes[0] / 8;  // x is [N, 8]

    // Workspace layout: ev[E] f32 | stats[N] int4 (sum, cnt, maxkey, minkey)
    char* ws    = (char*)d_ws;
    float* ev   = (float*)ws;
    int*   stats = (int*)(ws + (size_t)E * 4);

    init_nodes<<<(N + 255) / 256, 256, 0, stream>>>(stats, N);

    const int ntiles = E >> 4;
    if (ntiles > 0) {
        edge_mlp_scatter<<<2048, 256, 0, stream>>>(
            edge_attr, adjs, lin_e_w, lin_e_b, ev, stats, E, ntiles);
    }
    const int tail_start = ntiles << 4;
    const int rem = E - tail_start;
    if (rem > 0) {
        edge_tail<<<(rem + 255) / 256, 256, 0, stream>>>(
            edge_attr, adjs, lin_e_w, lin_e_b, ev, stats, tail_start, E);
    }

    finalize<<<(E + 255) / 256, 256, 0, stream>>>(
        ev, adjs, stats, aggr2_w, aggr2_b, lin_l_w, lin_l_b, out, E);
}